// ComposeLayerEncoder_41858751266842
// MI455X (gfx1250) — compile-verified
//
#include <hip/hip_runtime.h>
#include <hip/hip_bf16.h>

typedef __attribute__((ext_vector_type(2))) float v2f;
typedef __attribute__((ext_vector_type(8))) float v8f;

#define NBF 128   // NB_FEATS

// values pushed through these atomics are always >= 0 (post-relu), so the
// IEEE-754 bit pattern orders identically to unsigned int.
__device__ __forceinline__ void atomicMaxF(unsigned int* addr, float v) {
  atomicMax(addr, __float_as_uint(v));
}

// ---------------------------------------------------------------- init ----
__global__ void k_init(unsigned int* __restrict__ p, int n) {
  int i = blockIdx.x * blockDim.x + threadIdx.x;
  if (i < n) p[i] = 0u;
}

// ---- prep: padded conv weight Bpad[n][k] (128 x 16), bias folded at k=13 -
// Stored transposed (n-major) so a WMMA B operand is one aligned v2f load.
__global__ void k_prep_b(const float* __restrict__ convw,
                         const float* __restrict__ convb,
                         float* __restrict__ Bpad) {
  int i = blockIdx.x * blockDim.x + threadIdx.x;   // i = n*16 + k
  if (i >= NBF * 16) return;
  int n = i >> 4;
  int k = i & 15;
  float v = 0.f;
  if (k < 13)       v = convw[n * 13 + k];
  else if (k == 13) v = convb[n];
  Bpad[i] = v;
}

// ------------------------------------------- stage 1+2: MLP + segment max -
__global__ __launch_bounds__(256) void k_mlp_pool(
    const float* __restrict__ rp, const int* __restrict__ cluster,
    const float* __restrict__ w1, const float* __restrict__ b1,
    const float* __restrict__ w2, const float* __restrict__ b2,
    unsigned int* __restrict__ pooledU, int N, int C)
{
  int p = blockIdx.x * blockDim.x + threadIdx.x;
  if (p >= N) return;
  float x0 = rp[p*3+0], x1 = rp[p*3+1], x2 = rp[p*3+2];
  float h1[20];
  #pragma unroll
  for (int j = 0; j < 20; j++) {
    float s = __builtin_fmaf(w1[j*3+0], x0,
              __builtin_fmaf(w1[j*3+1], x1,
              __builtin_fmaf(w1[j*3+2], x2, b1[j])));
    h1[j] = fmaxf(s, 0.f);
  }
  float h2[10];
  #pragma unroll
  for (int j = 0; j < 10; j++) {
    float s = b2[j];
    #pragma unroll
    for (int k = 0; k < 20; k++) s = __builtin_fmaf(w2[j*20+k], h1[k], s);
    h2[j] = fmaxf(s, 0.f);
  }
  int c = cluster[p];
  c = min(max(c, 0), C - 1);
  int lane = threadIdx.x & 31;
  int waveBase = p - lane;
  bool fullwave = (waveBase + 31) < N;
  int cFirst = __shfl(c, 0, 32);
  int cLast  = __shfl(c, 31, 32);
  if (fullwave && cFirst == cLast) {
    // whole wave in one (sorted) cluster: tree-reduce, 1 atomic per dim
    #pragma unroll
    for (int d = 0; d < 10; d++) {
      float v = h2[d];
      v = fmaxf(v, __shfl_xor(v, 16, 32));
      v = fmaxf(v, __shfl_xor(v,  8, 32));
      v = fmaxf(v, __shfl_xor(v,  4, 32));
      v = fmaxf(v, __shfl_xor(v,  2, 32));
      v = fmaxf(v, __shfl_xor(v,  1, 32));
      if (lane == 0) atomicMaxF(&pooledU[c*10 + d], v);
    }
  } else {
    #pragma unroll
    for (int d = 0; d < 10; d++) atomicMaxF(&pooledU[c*10 + d], h2[d]);
  }
}

// ----------------------------------------------- stage 3: encoder FC 10->5
__global__ void k_enc_fc(const float* __restrict__ pooled,
                         const float* __restrict__ fw,
                         const float* __restrict__ fb,
                         float* __restrict__ encoded, int C)
{
  int c = blockIdx.x * blockDim.x + threadIdx.x;
  if (c >= C) return;
  float pv[10];
  #pragma unroll
  for (int k = 0; k < 10; k++) pv[k] = pooled[c*10 + k];
  #pragma unroll
  for (int j = 0; j < 5; j++) {
    float s = fb[j];
    #pragma unroll
    for (int k = 0; k < 10; k++) s = __builtin_fmaf(fw[j*10+k], pv[k], s);
    encoded[c*5 + j] = fmaxf(s, 0.f);
  }
}

// --------------- stage 4+5: fused concat @ conv_w^T (WMMA) + segment max -
// Block = 256 threads = 8 waves; each wave owns a 16-point tile (128 pts/blk).
// A tile: 16 pts x 16 K (K: 3 rel, 5 feat, 5 enc, 1.0 bias row, 2 zero pad).
// B tile: padded/transposed Bpad[n][k] -> one v2f load per WMMA operand.
__global__ __launch_bounds__(256) void k_conv_pool(
    const float* __restrict__ rp, const float* __restrict__ feat,
    const int* __restrict__ cluster, const float* __restrict__ encoded,
    const float* __restrict__ Bpad,
    unsigned int* __restrict__ featsU, int N, int C)
{
  __shared__ float sA[128][17];   // stride-17 pad: conflict-free WMMA gathers
  __shared__ int   sC[128];
  int tid  = threadIdx.x;
  int base = blockIdx.x * 128;

  if (tid < 128) {
    int p = base + tid;
    float row[16];
    #pragma unroll
    for (int k = 0; k < 16; k++) row[k] = 0.f;
    int c = 0;
    if (p < N) {
      row[0] = rp[p*3+0]; row[1] = rp[p*3+1]; row[2] = rp[p*3+2];
      #pragma unroll
      for (int k = 0; k < 5; k++) row[3+k] = feat[p*5 + k];
      c = cluster[p];
      c = min(max(c, 0), C - 1);
      #pragma unroll
      for (int k = 0; k < 5; k++) row[8+k] = encoded[c*5 + k];
      row[13] = 1.0f;             // bias row
    }
    #pragma unroll
    for (int k = 0; k < 16; k++) sA[tid][k] = row[k];
    sC[tid] = c;
  }
  __syncthreads();

  int lane = tid & 31;
  int wave = tid >> 5;
  int m16  = lane & 15;
  int hi   = lane >> 4;
  int rowBase = wave * 16;

  // fp32 WMMA A operand: lane holds row m16, K = 4q + 2*hi + {0,1}
  v2f a[4];
  #pragma unroll
  for (int q = 0; q < 4; q++) {
    a[q].x = sA[rowBase + m16][4*q + 2*hi + 0];
    a[q].y = sA[rowBase + m16][4*q + 2*hi + 1];
  }

  int tileP  = base + rowBase;
  bool uniform = (tileP + 15 < N) && (sC[rowBase] == sC[rowBase + 15]);
  int cUni = sC[rowBase];

  #pragma unroll
  for (int t = 0; t < 8; t++) {          // 8 column tiles of 16 features
    int n = t*16 + m16;
    const float* brow = Bpad + n * 16;   // Bpad[n][0..15], 8B aligned
    v2f b[4];
    #pragma unroll
    for (int q = 0; q < 4; q++)
      b[q] = *(const v2f*)(brow + 4*q + 2*hi);

    v8f acc = {0.f,0.f,0.f,0.f,0.f,0.f,0.f,0.f};
    #pragma unroll
    for (int q = 0; q < 4; q++)
      acc = __builtin_amdgcn_wmma_f32_16x16x4_f32(
                false, a[q], false, b[q], (short)0, acc, false, false);

    if (uniform) {
      // relu + reduce 16 rows to 1; one atomic per feature from lanes 0-15
      float v = 0.f;
      #pragma unroll
      for (int r = 0; r < 8; r++) v = fmaxf(v, acc[r]);
      v = fmaxf(v, __shfl_xor(v, 16, 32));
      if (lane < 16) atomicMaxF(&featsU[(size_t)cUni * NBF + n], v);
    } else {
      #pragma unroll
      for (int r = 0; r < 8; r++) {
        int m = r + 8*hi;
        int p = tileP + m;
        if (p < N) {
          int c = sC[rowBase + m];
          atomicMaxF(&featsU[(size_t)c * NBF + n], fmaxf(acc[r], 0.f));
        }
      }
    }
  }
}

// ------------------------ stage 6: out = relu(feats @ fc_w^T + fc_b), WMMA
// Wave per 16x16 output tile; K=128 as 32 chained fp32 WMMAs.
__global__ __launch_bounds__(256) void k_fc_out(
    const float* __restrict__ feats, const float* __restrict__ fcw,
    const float* __restrict__ fcb, float* __restrict__ out, int C)
{
  int tid  = threadIdx.x;
  int lane = tid & 31;
  int wave = tid >> 5;
  int gw   = blockIdx.x * 8 + wave;
  int rowtile = gw >> 3;               // C/16 row tiles
  int coltile = gw & 7;                // 128/16 col tiles
  int m16 = lane & 15;
  int hi  = lane >> 4;

  const float* arow = feats + (size_t)(rowtile*16 + m16) * NBF;
  const float* brow = fcw   + (size_t)(coltile*16 + m16) * NBF; // B=fc_w^T
  v8f acc = {0.f,0.f,0.f,0.f,0.f,0.f,0.f,0.f};
  #pragma unroll 8
  for (int q = 0; q < 32; q++) {
    int k0 = 4*q + 2*hi;               // 8B-aligned -> global_load_b64
    v2f av = *(const v2f*)(arow + k0);
    v2f bv = *(const v2f*)(brow + k0);
    acc = __builtin_amdgcn_wmma_f32_16x16x4_f32(
              false, av, false, bv, (short)0, acc, false, false);
  }
  int n = coltile*16 + m16;
  float bias = fcb[n];
  #pragma unroll
  for (int r = 0; r < 8; r++) {
    int row = rowtile*16 + r + 8*hi;
    out[(size_t)row * NBF + n] = fmaxf(acc[r] + bias, 0.f);
  }
}

// --------------------------------------------------------------- launch --
extern "C" void kernel_launch(void* const* d_in, const int* in_sizes, int n_in,
                              void* d_out, int out_size, void* d_ws, size_t ws_size,
                              hipStream_t stream) {
  const float* rel_pts  = (const float*)d_in[0];
  const float* features = (const float*)d_in[1];
  const int*   cluster  = (const int*)  d_in[2];
  const float* enc_w1   = (const float*)d_in[3];
  const float* enc_b1   = (const float*)d_in[4];
  const float* enc_w2   = (const float*)d_in[5];
  const float* enc_b2   = (const float*)d_in[6];
  const float* enc_fc_w = (const float*)d_in[7];
  const float* enc_fc_b = (const float*)d_in[8];
  const float* conv_w   = (const float*)d_in[9];
  const float* conv_b   = (const float*)d_in[10];
  const float* fc_w     = (const float*)d_in[11];
  const float* fc_b     = (const float*)d_in[12];
  float* out = (float*)d_out;

  const int N = in_sizes[0] / 3;       // 1,000,000
  const int C = out_size / NBF;        // 16,384

  // ws layout (floats): pooled[C*10] | encoded[C*5] | feats[C*128] | Bpad[2048]
  float* ws = (float*)d_ws;
  unsigned int* pooledU = (unsigned int*)ws;
  float*        pooled  = ws;
  float*        encoded = ws + (size_t)C * 10;
  unsigned int* featsU  = (unsigned int*)(ws + (size_t)C * 15);
  float*        feats   = (float*)featsU;
  float*        Bpad    = ws + (size_t)C * 143;

  int initN = C * 143;                 // zero pooled+encoded+feats region
  k_init<<<(initN + 255)/256, 256, 0, stream>>>((unsigned int*)ws, initN);

  k_prep_b<<<(NBF*16 + 255)/256, 256, 0, stream>>>(conv_w, conv_b, Bpad);

  k_mlp_pool<<<(N + 255)/256, 256, 0, stream>>>(
      rel_pts, cluster, enc_w1, enc_b1, enc_w2, enc_b2, pooledU, N, C);

  k_enc_fc<<<(C + 255)/256, 256, 0, stream>>>(
      pooled, enc_fc_w, enc_fc_b, encoded, C);

  k_conv_pool<<<(N + 127)/128, 256, 0, stream>>>(
      rel_pts, features, cluster, encoded, Bpad, featsU, N, C);

  // (C/16 row tiles) * 8 col tiles waves, 8 waves per block
  k_fc_out<<<(C/16) * 8 / 8, 256, 0, stream>>>(feats, fc_w, fc_b, out, C);
}